// Net_70248485093393
// MI455X (gfx1250) — compile-verified
//
#include <hip/hip_runtime.h>
#include <math.h>

#define NN 50000
#define EE 800000
#define FF 128
#define DIMM 128
#define NFAC 4
#define NDD 32
#define NLL 3
#define NCC 10
#define GG 256

typedef float v2f __attribute__((ext_vector_type(2)));
typedef float v8f __attribute__((ext_vector_type(8)));

static __device__ __forceinline__ v8f wmma4(v2f a, v2f b, v8f c) {
  return __builtin_amdgcn_wmma_f32_16x16x4_f32(false, a, false, b, (short)0, c, false, false);
}
static __device__ __forceinline__ float fsig(float x) { return 1.0f / (1.0f + __expf(-x)); }
static __device__ __forceinline__ float ftanh(float x) { return 1.0f - 2.0f / (1.0f + __expf(2.0f * x)); }

// ---------------- fold attention weights: vl[f] = lin0_W[f] @ Wl, etc.
__global__ void k_fold(const float* __restrict__ lin0W, const float* __restrict__ lin0b,
                       const float* __restrict__ attW, const float* __restrict__ attb,
                       float* vl, float* vr, float* ab) {
  int f = blockIdx.x, i = threadIdx.x;  // 4 x 128
  const float* W = lin0W + f * 16384;
  const float* wl = attW + f * 256;
  const float* wr = wl + 128;
  float sl = 0.f, sr = 0.f;
  for (int j = 0; j < 128; j++) { float wv = W[i * 128 + j]; sl += wv * wl[j]; sr += wv * wr[j]; }
  vl[f * 128 + i] = sl;
  vr[f * 128 + i] = sr;
  if (i == 0) {
    float bl = 0.f, br = 0.f;
    const float* b = lin0b + f * 128;
    for (int j = 0; j < 128; j++) { bl += b[j] * wl[j]; br += b[j] * wr[j]; }
    ab[f] = bl;
    ab[4 + f] = br + attb[f];
  }
}

// ---------------- pack enc_W (4,128,32) -> Wcat (128,128) row-major (K x N); zero accumulators
__global__ void k_pack(const float* __restrict__ encW, const float* __restrict__ encB,
                       float* Wcat, float* bcat, float* counts, float* pooled) {
  int idx = blockIdx.x * blockDim.x + threadIdx.x;
  if (idx < 16384) {
    int k = idx >> 7, n = idx & 127;
    Wcat[k * 128 + n] = encW[(n >> 5) * 4096 + k * 32 + (n & 31)];
  }
  if (idx < 128) bcat[idx] = encB[idx];
  if (idx < GG) counts[idx] = 0.f;
  if (idx < GG * 128) pooled[idx] = 0.f;
}

__global__ void k_counts(const int* __restrict__ batch, float* counts) {
  int i = blockIdx.x * blockDim.x + threadIdx.x;
  if (i < NN) atomicAdd(&counts[batch[i]], 1.0f);
}

__global__ void k_zero(float* p, int n) {
  int i = blockIdx.x * blockDim.x + threadIdx.x;
  if (i < n) p[i] = 0.f;
}

// ---------------- per-node attention projections al/ar (N x 4)
__global__ void k_nodeatt(const float* __restrict__ x, const float* __restrict__ vl,
                          const float* __restrict__ vr, const float* __restrict__ ab,
                          float* al, float* ar) {
  __shared__ float svl[512], svr[512], sab[8];
  int tid = threadIdx.x;  // 128
  for (int t = tid; t < 512; t += 128) { svl[t] = vl[t]; svr[t] = vr[t]; }
  if (tid < 8) sab[tid] = ab[tid];
  __syncthreads();
  int i = blockIdx.x * 128 + tid;
  if (i >= NN) return;
  float sl[4] = {0.f, 0.f, 0.f, 0.f}, sr[4] = {0.f, 0.f, 0.f, 0.f};
  for (int j = 0; j < 128; j++) {
    float xv = x[i * 128 + j];
#pragma unroll
    for (int f = 0; f < 4; f++) { sl[f] += xv * svl[f * 128 + j]; sr[f] += xv * svr[f * 128 + j]; }
  }
#pragma unroll
  for (int f = 0; f < 4; f++) {
    al[i * 4 + f] = sl[f] + sab[f];
    ar[i * 4 + f] = sr[f] + sab[4 + f];
  }
}

// ---------------- WMMA: out0 = x @ Wcat + bcat  (N x 128)
__global__ void k_encode(const float* __restrict__ x, const float* __restrict__ Wcat,
                         const float* __restrict__ bcat, float* __restrict__ out) {
  __shared__ float xs[64 * 132];
  int tid = threadIdx.x;  // 128 = 4 waves
  int base = blockIdx.x * 64;
  bool full = (base + 64 <= NN);  // block-uniform fast path (781/782 blocks)
  if (full) {
    for (int t = tid; t < 64 * 128; t += 128) {
      int r = t >> 7, c = t & 127;
      xs[r * 132 + c] = x[(base + r) * 128 + c];
    }
  } else {
    for (int t = tid; t < 64 * 128; t += 128) {
      int r = t >> 7, c = t & 127;
      int row = base + r;
      xs[r * 132 + c] = (row < NN) ? x[row * 128 + c] : 0.f;
    }
  }
  __syncthreads();
  int w = tid >> 5, lane = tid & 31, half = lane >> 4, col = lane & 15;
  int r16 = w * 16;
  const float* arow = &xs[(r16 + col) * 132];
  for (int nt = 0; nt < 8; nt++) {
    v8f acc = {};
#pragma unroll 8
    for (int s = 0; s < 32; s++) {
      int k0 = 4 * s + 2 * half;
      v2f a, b;
      a.x = arow[k0];
      a.y = arow[k0 + 1];
      b.x = Wcat[k0 * 128 + nt * 16 + col];
      b.y = Wcat[(k0 + 1) * 128 + nt * 16 + col];
      acc = wmma4(a, b, acc);
    }
    float bias = bcat[nt * 16 + col];
    if (full) {
#pragma unroll
      for (int i = 0; i < 8; i++) {
        int row = base + r16 + i + 8 * half;
        out[row * 128 + nt * 16 + col] = acc[i] + bias;
      }
    } else {
#pragma unroll
      for (int i = 0; i < 8; i++) {
        int row = base + r16 + i + 8 * half;
        if (row < NN) out[row * 128 + nt * 16 + col] = acc[i] + bias;
      }
    }
  }
}

// ---------------- per-edge attention -> atts output region
__global__ void k_edgeatt(const int* __restrict__ ei, const float* __restrict__ al,
                          const float* __restrict__ ar, float* __restrict__ atts) {
  int e = blockIdx.x * 256 + threadIdx.x;
  if (e >= EE) return;
  int s = ei[e], d = ei[EE + e];
  float4 a4 = ((const float4*)al)[s];
  float4 b4 = ((const float4*)ar)[d];
  atts[0 * EE + e] = fsig(6.0f * (a4.x + b4.x));
  atts[1 * EE + e] = fsig(6.0f * (a4.y + b4.y));
  atts[2 * EE + e] = fsig(6.0f * (a4.z + b4.z));
  atts[3 * EE + e] = fsig(6.0f * (a4.w + b4.w));
}

// ---------------- edge scatter: agg[dst] += out[src] * att  (128 threads = 128 channels)
// EE is an exact multiple of 32, so the inner loop needs no bounds check.
__global__ void k_scatter(const int* __restrict__ ei, const float* __restrict__ out,
                          const float* __restrict__ atts, float* __restrict__ agg) {
  int tid = threadIdx.x;  // 0..127 = f*32+c
  int f = tid >> 5;
  int ebase = blockIdx.x * 32;
#pragma unroll 4
  for (int k = 0; k < 32; k++) {
    int e = ebase + k;
    int s = ei[e], d = ei[EE + e];
    float v = out[s * 128 + tid] * atts[f * EE + e];
    atomicAdd(&agg[d * 128 + tid], v);
  }
}

// ---------------- WMMA node update: conv + GRU, in-place on out. grid=(tiles, factor)
__global__ void k_update(const float* __restrict__ agg, float* out,
                         const float* __restrict__ Wrel, const float* __restrict__ brel,
                         const float* __restrict__ Wroot, const float* __restrict__ Wih,
                         const float* __restrict__ Whh, const float* __restrict__ bih,
                         const float* __restrict__ bhh, int layer) {
  __shared__ float sWrel[1024], sWroot[1024];
  __shared__ float sWihT[32 * 96], sWhhT[32 * 96];
  __shared__ float sBrel[32], sBih[96], sBhh[96];
  __shared__ float sA[4][16 * 36];  // agg tiles (A-operand, row-major)
  __shared__ float sH[4][16 * 36];  // h (= current out) tiles
  __shared__ float sM[4][16 * 36];  // m tiles
  int f = blockIdx.y;
  int tid = threadIdx.x;  // 128 = 4 waves
  const float* wrel = Wrel + (f * NLL + layer) * 1024;
  const float* wroot = Wroot + (f * NLL + layer) * 1024;
  for (int t = tid; t < 1024; t += 128) { sWrel[t] = wrel[t]; sWroot[t] = wroot[t]; }
  const float* wih = Wih + f * 3072;
  const float* whh = Whh + f * 3072;
  for (int t = tid; t < 3072; t += 128) {  // transpose (96,32) -> (32,96)
    int k = t & 31, n = t >> 5;
    sWihT[k * 96 + n] = wih[t];
    sWhhT[k * 96 + n] = whh[t];
  }
  if (tid < 32) sBrel[tid] = brel[(f * NLL + layer) * 32 + tid];
  if (tid < 96) { sBih[tid] = bih[f * 96 + tid]; sBhh[tid] = bhh[f * 96 + tid]; }

  int w = tid >> 5, lane = tid & 31;
  int rowbase = blockIdx.x * 64 + w * 16;
  bool full = (rowbase + 16 <= NN);  // wave-uniform
  if (full) {
    for (int t = lane; t < 512; t += 32) {
      int r = t >> 5, c = t & 31;
      sA[w][r * 36 + c] = agg[(rowbase + r) * 128 + f * 32 + c];
      sH[w][r * 36 + c] = out[(rowbase + r) * 128 + f * 32 + c];
    }
  } else {
    for (int t = lane; t < 512; t += 32) {
      int r = t >> 5, c = t & 31;
      int row = rowbase + r;
      float av = 0.f, hv = 0.f;
      if (row < NN) { av = agg[row * 128 + f * 32 + c]; hv = out[row * 128 + f * 32 + c]; }
      sA[w][r * 36 + c] = av;
      sH[w][r * 36 + c] = hv;
    }
  }
  __syncthreads();

  int half = lane >> 4, col = lane & 15;
  const float* arowA = &sA[w][col * 36];
  const float* arowH = &sH[w][col * 36];
  const float* arowM = &sM[w][col * 36];

  // m = relu(agg@Wrel + out@Wroot + brel)  -> two 16x16 tiles, staged to sM
  for (int nt = 0; nt < 2; nt++) {
    v8f acc = {};
#pragma unroll
    for (int s = 0; s < 8; s++) {
      int k0 = 4 * s + 2 * half;
      v2f a, b;
      a.x = arowA[k0]; a.y = arowA[k0 + 1];
      b.x = sWrel[k0 * 32 + nt * 16 + col]; b.y = sWrel[(k0 + 1) * 32 + nt * 16 + col];
      acc = wmma4(a, b, acc);
      a.x = arowH[k0]; a.y = arowH[k0 + 1];
      b.x = sWroot[k0 * 32 + nt * 16 + col]; b.y = sWroot[(k0 + 1) * 32 + nt * 16 + col];
      acc = wmma4(a, b, acc);
    }
    float bias = sBrel[nt * 16 + col];
#pragma unroll
    for (int i = 0; i < 8; i++) {
      float v = acc[i] + bias;
      sM[w][(i + 8 * half) * 36 + nt * 16 + col] = v > 0.f ? v : 0.f;
    }
  }
  __syncthreads();

  // gates r,z: rz = sigmoid(m@WihT + h@WhhT + bih + bhh), channels 0..63 (4 tiles)
  v8f rz[4];
  for (int nt = 0; nt < 4; nt++) {
    v8f acc = {};
#pragma unroll
    for (int s = 0; s < 8; s++) {
      int k0 = 4 * s + 2 * half;
      v2f a, b;
      a.x = arowM[k0]; a.y = arowM[k0 + 1];
      b.x = sWihT[k0 * 96 + nt * 16 + col]; b.y = sWihT[(k0 + 1) * 96 + nt * 16 + col];
      acc = wmma4(a, b, acc);
      a.x = arowH[k0]; a.y = arowH[k0 + 1];
      b.x = sWhhT[k0 * 96 + nt * 16 + col]; b.y = sWhhT[(k0 + 1) * 96 + nt * 16 + col];
      acc = wmma4(a, b, acc);
    }
    int ch = nt * 16 + col;
    float bb = sBih[ch] + sBhh[ch];
#pragma unroll
    for (int i = 0; i < 8; i++) rz[nt][i] = fsig(acc[i] + bb);
  }

  // n gate + h update, channels 64..95 viewed as 2 tiles aligned with rz tiles
  for (int nt = 0; nt < 2; nt++) {
    v8f accI = {}, accH = {};
#pragma unroll
    for (int s = 0; s < 8; s++) {
      int k0 = 4 * s + 2 * half;
      v2f a, b;
      a.x = arowM[k0]; a.y = arowM[k0 + 1];
      b.x = sWihT[k0 * 96 + 64 + nt * 16 + col]; b.y = sWihT[(k0 + 1) * 96 + 64 + nt * 16 + col];
      accI = wmma4(a, b, accI);
      a.x = arowH[k0]; a.y = arowH[k0 + 1];
      b.x = sWhhT[k0 * 96 + 64 + nt * 16 + col]; b.y = sWhhT[(k0 + 1) * 96 + 64 + nt * 16 + col];
      accH = wmma4(a, b, accH);
    }
    int ch = 64 + nt * 16 + col;
    float bI = sBih[ch], bH = sBhh[ch];
#pragma unroll
    for (int i = 0; i < 8; i++) {
      float hn = accH[i] + bH;
      float nn = ftanh(accI[i] + bI + rz[nt][i] * hn);
      int rrow = i + 8 * half;
      float hprev = sH[w][rrow * 36 + nt * 16 + col];
      float z = rz[2 + nt][i];
      float hv = (1.0f - z) * nn + z * hprev;
      int row = rowbase + rrow;
      if (full) {
        out[row * 128 + f * 32 + nt * 16 + col] = hv;
      } else if (row < NN) {
        out[row * 128 + f * 32 + nt * 16 + col] = hv;
      }
    }
  }
}

// ---------------- pooled mean per graph
__global__ void k_pool(const float* __restrict__ out, const int* __restrict__ batch,
                       const float* __restrict__ counts, float* pooled) {
  int idx = blockIdx.x * 256 + threadIdx.x;
  int i = idx >> 7, ch = idx & 127;
  if (i >= NN) return;
  int g = batch[i];
  float inv = 1.0f / fmaxf(counts[g], 1.0f);
  atomicAdd(&pooled[g * 128 + ch], out[idx] * inv);
}

__global__ void k_outs(const float* __restrict__ pooled, float* __restrict__ outs) {
  int idx = blockIdx.x * 256 + threadIdx.x;
  if (idx >= NFAC * GG * NDD) return;
  int f = idx / (GG * NDD);
  int rem = idx - f * (GG * NDD);
  int g = rem / NDD, c = rem - g * NDD;
  outs[idx] = pooled[g * 128 + f * 32 + c];
}

// ---------------- final MLP: pred = relu(pooled @ fc1 + b1) @ fc2 + b2
__global__ void k_mlp(const float* __restrict__ pooled, const float* __restrict__ fc1W,
                      const float* __restrict__ fc1b, const float* __restrict__ fc2W,
                      const float* __restrict__ fc2b, float* __restrict__ pred) {
  __shared__ float row[128], hid[128];
  int g = blockIdx.x, tid = threadIdx.x;  // 256 blocks x 128
  row[tid] = pooled[g * 128 + tid];
  __syncthreads();
  float h = fc1b[tid];
  for (int i = 0; i < 128; i++) h += row[i] * fc1W[i * 128 + tid];
  hid[tid] = h > 0.f ? h : 0.f;
  __syncthreads();
  if (tid < NCC) {
    float p = fc2b[tid];
    for (int j = 0; j < 128; j++) p += hid[j] * fc2W[j * NCC + tid];
    pred[g * NCC + tid] = p;
  }
}

extern "C" void kernel_launch(void* const* d_in, const int* in_sizes, int n_in,
                              void* d_out, int out_size, void* d_ws, size_t ws_size,
                              hipStream_t stream) {
  const float* x = (const float*)d_in[0];
  const int* ei = (const int*)d_in[1];
  const int* batch = (const int*)d_in[2];
  const float* lin0W = (const float*)d_in[3];
  const float* lin0b = (const float*)d_in[4];
  const float* attW = (const float*)d_in[5];
  const float* attb = (const float*)d_in[6];
  const float* encW = (const float*)d_in[7];
  const float* encB = (const float*)d_in[8];
  const float* Wrel = (const float*)d_in[9];
  const float* brel = (const float*)d_in[10];
  const float* Wroot = (const float*)d_in[11];
  const float* Wih = (const float*)d_in[12];
  const float* Whh = (const float*)d_in[13];
  const float* bih = (const float*)d_in[14];
  const float* bhh = (const float*)d_in[15];
  const float* fc1W = (const float*)d_in[16];
  const float* fc1b = (const float*)d_in[17];
  const float* fc2W = (const float*)d_in[18];
  const float* fc2b = (const float*)d_in[19];

  float* pred = (float*)d_out;
  float* atts = pred + GG * NCC;           // 2560
  float* outs = atts + (size_t)NFAC * EE;  // + 3.2M

  float* ws = (float*)d_ws;
  float* vl = ws;                 // 512
  float* vr = ws + 512;           // 512
  float* ab = ws + 1024;          // 8
  float* bcat = ws + 1032;        // 128
  float* Wcat = ws + 1160;        // 16384
  float* counts = ws + 17544;     // 256
  float* pooled = ws + 17800;     // 32768
  float* al = ws + 50568;         // 200000 (16B aligned)
  float* ar = ws + 250568;        // 200000 (16B aligned)
  float* outbuf = ws + 450568;    // 6.4M
  float* agg = ws + 6850568;      // 6.4M

  k_fold<<<NFAC, 128, 0, stream>>>(lin0W, lin0b, attW, attb, vl, vr, ab);
  k_pack<<<128, 256, 0, stream>>>(encW, encB, Wcat, bcat, counts, pooled);
  k_counts<<<(NN + 255) / 256, 256, 0, stream>>>(batch, counts);
  k_nodeatt<<<(NN + 127) / 128, 128, 0, stream>>>(x, vl, vr, ab, al, ar);
  k_encode<<<(NN + 63) / 64, 128, 0, stream>>>(x, Wcat, bcat, outbuf);
  k_edgeatt<<<(EE + 255) / 256, 256, 0, stream>>>(ei, al, ar, atts);

  for (int l = 0; l < NLL; l++) {
    k_zero<<<(NN * 128 + 255) / 256, 256, 0, stream>>>(agg, NN * 128);
    k_scatter<<<(EE + 31) / 32, 128, 0, stream>>>(ei, outbuf, atts, agg);
    k_update<<<dim3((NN + 63) / 64, NFAC), 128, 0, stream>>>(agg, outbuf, Wrel, brel, Wroot,
                                                             Wih, Whh, bih, bhh, l);
  }

  k_pool<<<(NN * 128 + 255) / 256, 256, 0, stream>>>(outbuf, batch, counts, pooled);
  k_outs<<<(NFAC * GG * NDD + 255) / 256, 256, 0, stream>>>(pooled, outs);
  k_mlp<<<GG, 128, 0, stream>>>(pooled, fc1W, fc1b, fc2W, fc2b, pred);
}